// SAGE_29162827939935
// MI455X (gfx1250) — compile-verified
//
#include <hip/hip_runtime.h>

#define N_NODES 10000
#define N_EDGES 640000
#define IN_F    128
#define HID_F   256
#define OUT_F   128

typedef float v2f __attribute__((ext_vector_type(2)));
typedef float v8f __attribute__((ext_vector_type(8)));

// ---------------------------------------------------------------------------
// Utility: zero a float region
// ---------------------------------------------------------------------------
__global__ __launch_bounds__(256) void sage_zero_kernel(float* __restrict__ p, int n) {
  int i = blockIdx.x * blockDim.x + threadIdx.x;
  if (i < n) p[i] = 0.0f;
}

// ---------------------------------------------------------------------------
// In-degree via float atomics (reference computes degree as float segment_sum)
// ---------------------------------------------------------------------------
__global__ __launch_bounds__(256) void sage_degree_kernel(const int* __restrict__ dst,
                                                          float* __restrict__ deg) {
  int e = blockIdx.x * blockDim.x + threadIdx.x;
  if (e < N_EDGES) atomicAdd(&deg[dst[e]], 1.0f);
}

__global__ __launch_bounds__(256) void sage_rdeg_kernel(const float* __restrict__ deg,
                                                        float* __restrict__ rdeg) {
  int i = blockIdx.x * blockDim.x + threadIdx.x;
  if (i < N_NODES) rdeg[i] = 1.0f / fmaxf(deg[i], 1.0f);
}

// ---------------------------------------------------------------------------
// Edge-parallel scatter-add: agg[dst[e], :] += h[src[e], :]
// One thread handles one (edge, 4-float chunk). F/4 threads per edge,
// consecutive threads cover consecutive chunks -> coalesced gather & scatter.
// Atomics land in L2 (node features ~5-10MB << 192MB L2).
// ---------------------------------------------------------------------------
template <int F>
__global__ __launch_bounds__(256) void sage_scatter_kernel(const float* __restrict__ h,
                                                           const int* __restrict__ src,
                                                           const int* __restrict__ dst,
                                                           float* __restrict__ agg) {
  constexpr int CHUNKS = F / 4;            // threads per edge (32 or 64)
  constexpr int EPB    = 256 / CHUNKS;     // edges per block
  const int e = blockIdx.x * EPB + (threadIdx.x / CHUNKS);
  const int c = threadIdx.x & (CHUNKS - 1);
  if (e >= N_EDGES) return;
  const int s = src[e];
  const int d = dst[e];
  const float4 v = *(const float4*)(h + (long long)s * F + c * 4);
  float* o = agg + (long long)d * F + c * 4;
  atomicAdd(o + 0, v.x);
  atomicAdd(o + 1, v.y);
  atomicAdd(o + 2, v.z);
  atomicAdd(o + 3, v.w);
}

// ---------------------------------------------------------------------------
// Fused SAGE layer via V_WMMA_F32_16X16X4_F32 (wave32, one wave per 16x16 tile)
//   Out = act( Hs @ Wself + (Agg * rdeg[row]) @ Wneigh + bias )
//
// A 16x4 f32 layout : lane&15 = M row; VGPR pair = K = 2*(lane>>4), +1
// B 4x16  f32 layout: lane&15 = N col; VGPR pair = K = 2*(lane>>4), +1
// C/D 16x16 layout  : VGPR v -> M = v + 8*(lane>>4); lane&15 = N col
// ---------------------------------------------------------------------------
template <int K, int NCOLS, bool RELU>
__global__ __launch_bounds__(256) void sage_gemm_wmma(const float* __restrict__ Hs,
                                                      const float* __restrict__ Agg,
                                                      const float* __restrict__ rdeg,
                                                      const float* __restrict__ Wself,
                                                      const float* __restrict__ Wneigh,
                                                      const float* __restrict__ bias,
                                                      float* __restrict__ Out) {
  const int lane = threadIdx.x & 31;
  const int wave = threadIdx.x >> 5;
  const int tile = blockIdx.x * 8 + wave;       // 8 waves per 256-thread block
  constexpr int COL_TILES = NCOLS / 16;
  const int tr = tile / COL_TILES;              // row tile  (N_NODES/16 = 625 exact)
  const int tc = tile % COL_TILES;              // col tile

  const int arow_idx = tr * 16 + (lane & 15);   // A-matrix row for this lane
  const int col      = tc * 16 + (lane & 15);   // B/C/D column for this lane
  const int koff     = (lane >> 4) * 2;         // K sub-offset per half-wave

  const float rd = rdeg[arow_idx];
  const float* __restrict__ hrow = Hs  + (long long)arow_idx * K;
  const float* __restrict__ grow = Agg + (long long)arow_idx * K;

  v8f acc = {0.f, 0.f, 0.f, 0.f, 0.f, 0.f, 0.f, 0.f};

  // Self term: Hs @ Wself
#pragma unroll 4
  for (int kb = 0; kb < K; kb += 4) {
    v2f a, b;
    a[0] = hrow[kb + koff];
    a[1] = hrow[kb + koff + 1];
    b[0] = Wself[(kb + koff) * NCOLS + col];
    b[1] = Wself[(kb + koff + 1) * NCOLS + col];
    acc = __builtin_amdgcn_wmma_f32_16x16x4_f32(false, a, false, b,
                                                (short)0, acc, false, false);
  }

  // Neighbor term: (Agg / deg) @ Wneigh  (degree normalization fused here)
#pragma unroll 4
  for (int kb = 0; kb < K; kb += 4) {
    v2f a, b;
    a[0] = grow[kb + koff] * rd;
    a[1] = grow[kb + koff + 1] * rd;
    b[0] = Wneigh[(kb + koff) * NCOLS + col];
    b[1] = Wneigh[(kb + koff + 1) * NCOLS + col];
    acc = __builtin_amdgcn_wmma_f32_16x16x4_f32(false, a, false, b,
                                                (short)0, acc, false, false);
  }

  // Epilogue: bias (+ReLU), store 16x16 tile
  const float bv    = bias[col];
  const int   rbase = tr * 16 + ((lane >> 4) << 3);
#pragma unroll
  for (int v = 0; v < 8; ++v) {
    float o = acc[v] + bv;
    if (RELU) o = fmaxf(o, 0.0f);
    Out[(long long)(rbase + v) * NCOLS + col] = o;
  }
}

// ---------------------------------------------------------------------------
// Host-side orchestration
// ---------------------------------------------------------------------------
extern "C" void kernel_launch(void* const* d_in, const int* in_sizes, int n_in,
                              void* d_out, int out_size, void* d_ws, size_t ws_size,
                              hipStream_t stream) {
  const float* x        = (const float*)d_in[0];
  const float* w_self1  = (const float*)d_in[1];
  const float* w_neigh1 = (const float*)d_in[2];
  const float* b1       = (const float*)d_in[3];
  const float* w_self2  = (const float*)d_in[4];
  const float* w_neigh2 = (const float*)d_in[5];
  const float* b2       = (const float*)d_in[6];
  const int*   src      = (const int*)d_in[7];
  const int*   dst      = (const int*)d_in[8];
  float*       out      = (float*)d_out;

  // Workspace layout (floats)
  float* ws   = (float*)d_ws;
  float* deg  = ws;                         // [N_NODES]
  float* rdeg = ws + N_NODES;               // [N_NODES]
  float* agg  = ws + 2 * N_NODES;           // [N_NODES, HID_F] (reused both layers)
  float* h1   = agg + (size_t)N_NODES * HID_F;  // [N_NODES, HID_F]

  // 1. Zero deg + agg(layer-1 region, IN_F wide); deg/rdeg/agg are contiguous
  {
    const int n = 2 * N_NODES + N_NODES * IN_F;
    sage_zero_kernel<<<(n + 255) / 256, 256, 0, stream>>>(ws, n);
  }
  // 2. In-degree
  sage_degree_kernel<<<(N_EDGES + 255) / 256, 256, 0, stream>>>(dst, deg);
  // 3. Reciprocal degree
  sage_rdeg_kernel<<<(N_NODES + 255) / 256, 256, 0, stream>>>(deg, rdeg);
  // 4. Layer-1 aggregation: agg += x[src] scattered to dst
  sage_scatter_kernel<IN_F><<<N_EDGES / (256 / (IN_F / 4)), 256, 0, stream>>>(x, src, dst, agg);
  // 5. Layer-1 fused GEMM + bias + ReLU -> h1 [N_NODES, HID_F]
  {
    const int tiles = (N_NODES / 16) * (HID_F / 16);   // 625*16 = 10000
    sage_gemm_wmma<IN_F, HID_F, true><<<tiles / 8, 256, 0, stream>>>(
        x, agg, rdeg, w_self1, w_neigh1, b1, h1);
  }
  // 6. Re-zero agg (HID_F wide now)
  {
    const int n = N_NODES * HID_F;
    sage_zero_kernel<<<(n + 255) / 256, 256, 0, stream>>>(agg, n);
  }
  // 7. Layer-2 aggregation: agg += h1[src] scattered to dst
  sage_scatter_kernel<HID_F><<<N_EDGES / (256 / (HID_F / 4)), 256, 0, stream>>>(h1, src, dst, agg);
  // 8. Layer-2 fused GEMM + bias -> d_out [N_NODES, OUT_F]
  {
    const int tiles = (N_NODES / 16) * (OUT_F / 16);   // 625*8 = 5000
    sage_gemm_wmma<HID_F, OUT_F, false><<<tiles / 8, 256, 0, stream>>>(
        h1, agg, rdeg, w_self2, w_neigh2, b2, out);
  }
}